// LuongsAttention_19834158973688
// MI455X (gfx1250) — compile-verified
//
#include <hip/hip_runtime.h>
#include <math.h>
#include <stdint.h>

typedef __attribute__((ext_vector_type(2))) float v2f;
typedef __attribute__((ext_vector_type(8))) float v8f;
typedef __attribute__((ext_vector_type(4))) unsigned int v4u;
typedef __attribute__((ext_vector_type(4))) int v4i;
typedef __attribute__((ext_vector_type(8))) int v8i;

#define B_DIM 32
#define S_DIM 2048
#define U_DIM 1024

#define TC      16                // u-columns per tile / block
#define THREADS 256
#define CPT     (THREADS / TC)    // 16 chunk-threads per column
#define KITER   (S_DIM / CPT)     // 128 s-values per thread (s = c mod 16)

// ---------------------------------------------------------------------------
// Kernel 1: proj = H @ W + bias  via V_WMMA_F32_16X16X4_F32
// One wave per 16x16 output tile, grid (64, 2). proj is stashed in the
// context region of d_out; each (b,u) slot is read only by the block in
// kernel 2 that later overwrites it with the final context (barrier-ordered).
// ---------------------------------------------------------------------------
__global__ void __launch_bounds__(32)
LuongsAttention_proj_wmma(const float* __restrict__ H,
                          const float* __restrict__ W,
                          const float* __restrict__ bias,
                          float* __restrict__ proj_out) {
    const int lane  = threadIdx.x;
    const int l15   = lane & 15;
    const int half  = lane >> 4;
    const int nbase = blockIdx.x * 16;
    const int mbase = blockIdx.y * 16;

    v8f acc = {};
    const float* __restrict__ Arow = H + (size_t)(mbase + l15) * U_DIM;

    for (int k0 = 0; k0 < U_DIM; k0 += 4) {
        const int ka = k0 + half * 2;
        v2f a;
        a.x = Arow[ka];
        a.y = Arow[ka + 1];
        v2f b;
        b.x = W[(size_t)ka       * U_DIM + nbase + l15];
        b.y = W[(size_t)(ka + 1) * U_DIM + nbase + l15];
        acc = __builtin_amdgcn_wmma_f32_16x16x4_f32(
            false, a, false, b, (short)0, acc, false, false);
    }

    const int col = nbase + l15;
    const float bv = bias[col];
    const int rowbase = mbase + half * 8;
#pragma unroll
    for (int i = 0; i < 8; ++i) {
        proj_out[(size_t)(rowbase + i) * U_DIM + col] = acc[i] + bv;
    }
}

// ---------------------------------------------------------------------------
// Kernel 2: softmax over s + context, single HBM read of EO via TDM to LDS.
// Block = (b, 16-column slab). 128 KB LDS tile -> 2 blocks co-resident per
// WGP so one block's tensor DMA overlaps another's compute/stores.
// ---------------------------------------------------------------------------
__global__ void __launch_bounds__(256)
LuongsAttention_softmax_ctx(const float* __restrict__ EO,
                            float* __restrict__ out) {
    __shared__ float tile[S_DIM * TC];   // 128 KB: EO[b, :, u0:u0+16]
    __shared__ float red1[THREADS];
    __shared__ float red2[THREADS];
    __shared__ float mfin[TC];
    __shared__ float linv[TC];

    const int t  = threadIdx.x;
    const int uc = t & (TC - 1);     // column within slab
    const int c  = t >> 4;           // chunk thread: owns s == c (mod 16)
    const int u0 = blockIdx.x * TC;
    const int b  = blockIdx.y;

    float* __restrict__ ctx = out;                           // [B, U]
    float* __restrict__ aw  = out + (size_t)B_DIM * U_DIM;   // [B, S, U]

    const float* __restrict__ eobase = EO + (size_t)b * S_DIM * U_DIM + u0;
    const float p = ctx[(size_t)b * U_DIM + u0 + uc];        // proj (kernel 1)

#if __has_builtin(__builtin_amdgcn_tensor_load_to_lds) && \
    __has_builtin(__builtin_amdgcn_s_wait_tensorcnt)
    // --- Tensor Data Mover: DMA the 2D tile [2048 x 16] f32 into LDS ---
    if (t < 32) {   // wave 0 only (EXEC ignored by TDM; one issue per wave)
        const unsigned long long ga = (unsigned long long)(uintptr_t)eobase;
        const unsigned int lds_off = (unsigned int)(uintptr_t)&tile[0];
        v4u g0;
        g0[0] = 1u;                                     // count=1, user mode
        g0[1] = lds_off;                                // lds_addr (bytes)
        g0[2] = (unsigned int)ga;                       // global_addr[31:0]
        g0[3] = (unsigned int)((ga >> 32) & 0x1FFFFFFull) // global_addr[56:32]
                | 0x80000000u;                          // type=2 ("image")
        v8i g1;
        g1[0] = 0x00020000;                 // wg_mask=0, data_size=2 (4B)
        g1[1] = (int)((unsigned)U_DIM << 16);   // tensor_dim0[15:0]=1024
        g1[2] = (int)((unsigned)S_DIM << 16);   // dim0[31:16]=0, tensor_dim1[15:0]=2048
        g1[3] = (int)((unsigned)TC << 16);      // tensor_dim1[31:16]=0, tile_dim0=16
        g1[4] = S_DIM;                          // tile_dim1=2048, tile_dim2=0
        g1[5] = U_DIM;                          // tensor_dim0_stride[31:0]=1024
        g1[6] = 0;                              // stride0[47:32]=0, stride1 lo=0
        g1[7] = 0;
        const v4i z4 = {0, 0, 0, 0};            // groups 2/3 unused (2D)
        const v8i z8 = {0, 0, 0, 0, 0, 0, 0, 0};
        // 6-arg form (this toolchain): (g0, g1, g2, g3, extra, cpol)
        __builtin_amdgcn_tensor_load_to_lds(g0, g1, z4, z4, z8, 0);
        __builtin_amdgcn_s_wait_tensorcnt(0);
    }
    __syncthreads();
#else
    // Fallback: cooperative vectorized copy (still a single EO read)
    for (int idx = t; idx < (S_DIM * TC) / 4; idx += THREADS) {
        const int row = idx >> 2;
        const int q   = (idx & 3) * 4;
        const float4 v = *(const float4*)(eobase + (size_t)row * U_DIM + q);
        *(float4*)&tile[row * TC + q] = v;
    }
    __syncthreads();
#endif

    // ---- Pass 1a: per-chunk max of score = p * e ----
    float mloc = -INFINITY;
#pragma unroll 8
    for (int k = 0; k < KITER; ++k) {
        const int s = c + CPT * k;
        mloc = fmaxf(mloc, p * tile[s * TC + uc]);
    }
    red1[t] = mloc;
    __syncthreads();
    if (c == 0) {
        float m = red1[uc];
#pragma unroll
        for (int j = 1; j < CPT; ++j) m = fmaxf(m, red1[j * TC + uc]);
        mfin[uc] = m;
    }
    __syncthreads();
    const float m = mfin[uc];

    // ---- Pass 1b: one exp/element; stash w in-place, accumulate l and w*e ----
    float l = 0.0f, cv = 0.0f;
#pragma unroll 4
    for (int k = 0; k < KITER; ++k) {
        const int s = c + CPT * k;
        const float e = tile[s * TC + uc];
        const float w = __expf(fmaf(p, e, -m));
        tile[s * TC + uc] = w;
        l += w;
        cv = fmaf(w, e, cv);
    }
    red1[t] = l;
    red2[t] = cv;
    __syncthreads();
    if (c == 0) {
        float ls = 0.0f, cs = 0.0f;
#pragma unroll
        for (int j = 0; j < CPT; ++j) {
            ls += red1[j * TC + uc];
            cs += red2[j * TC + uc];
        }
        const float inv = 1.0f / ls;
        linv[uc] = inv;
        ctx[(size_t)b * U_DIM + u0 + uc] = cs * inv;   // overwrites proj stash
    }
    __syncthreads();
    const float inv = linv[uc];

    // ---- Pass 2: normalized weights out (non-temporal, never re-read) ----
    float* __restrict__ awp = aw + (size_t)b * S_DIM * U_DIM + u0 + uc;
#pragma unroll 4
    for (int k = 0; k < KITER; ++k) {
        const int s = c + CPT * k;
        __builtin_nontemporal_store(tile[s * TC + uc] * inv,
                                    &awp[(size_t)s * U_DIM]);
    }
}

// ---------------------------------------------------------------------------
extern "C" void kernel_launch(void* const* d_in, const int* in_sizes, int n_in,
                              void* d_out, int out_size, void* d_ws, size_t ws_size,
                              hipStream_t stream) {
    const float* H  = (const float*)d_in[0];   // [B, U]
    const float* EO = (const float*)d_in[1];   // [B, S, U]
    const float* Wk = (const float*)d_in[2];   // [U, U]
    const float* Wb = (const float*)d_in[3];   // [U]
    float* out = (float*)d_out;                // [B*U] context ++ [B*S*U] weights

    LuongsAttention_proj_wmma<<<dim3(U_DIM / 16, B_DIM / 16), 32, 0, stream>>>(
        H, Wk, Wb, out);

    LuongsAttention_softmax_ctx<<<dim3(U_DIM / TC, B_DIM), THREADS, 0, stream>>>(
        EO, out);
}